// ModelSpectral_32968168964273
// MI455X (gfx1250) — compile-verified
//
#include <hip/hip_runtime.h>
#include <math.h>

typedef __attribute__((ext_vector_type(2))) float v2f;
typedef __attribute__((ext_vector_type(8))) float v8f;

#define LW 32            // hidden width
#define WPB 8            // waves per block (256 threads, wave32)

// ---------------------------------------------------------------------------
// Coarse 2-node SAGE: x = eye(2), deg = 1, mean[i] = e_{1-i}
//   X[i,c] = tanh(Wc_l[1-i,c] + Wc_r[i,c] + bc[c])
// ---------------------------------------------------------------------------
__global__ void coarse_kernel(const float* __restrict__ Wc_l,
                              const float* __restrict__ Wc_r,
                              const float* __restrict__ bc,
                              float* __restrict__ X) {
  int t = threadIdx.x;
  if (t < 64) {
    int i = t >> 5;
    int c = t & 31;
    X[i * LW + c] = tanhf(Wc_l[(1 - i) * LW + c] + Wc_r[i * LW + c] + bc[c]);
  }
}

__global__ void zero_kernel(float* __restrict__ p, int count) {
  int i = blockIdx.x * blockDim.x + threadIdx.x;
  if (i < count) p[i] = 0.0f;
}

// Unpool gather: Y[i,:] = X[inv[i],:]   (8 float4 groups per node)
__global__ void unpool_kernel(float* __restrict__ Y, const float* __restrict__ X,
                              const int* __restrict__ inv, int n) {
  int tid = blockIdx.x * blockDim.x + threadIdx.x;
  if (tid >= n * 8) return;
  int i = tid >> 3, g = tid & 7;
  *(float4*)(Y + (size_t)i * LW + g * 4) =
      *(const float4*)(X + (size_t)inv[i] * LW + g * 4);
}

// Edge-parallel mean aggregation: MSG[dst] += X[src]; DEG[dst] += 1
// 8 threads per edge, each owns a float4 channel group.
__global__ void aggregate_kernel(const float* __restrict__ X,
                                 const int* __restrict__ src,
                                 const int* __restrict__ dst,
                                 float* __restrict__ MSG,
                                 float* __restrict__ DEG, int E) {
  int tid = blockIdx.x * blockDim.x + threadIdx.x;
  if (tid >= E * 8) return;
  int e = tid >> 3, g = tid & 7;
  int s = src[e], d = dst[e];
  // gfx1250 speculative prefetch of an upcoming source row (global_prefetch_b8)
  int e2 = e + 4096;
  if (g == 0 && e2 < E) __builtin_prefetch(X + (size_t)src[e2] * LW, 0, 1);
  float4 v = *(const float4*)(X + (size_t)s * LW + g * 4);
  float* out = MSG + (size_t)d * LW + g * 4;
  atomicAdd(out + 0, v.x);
  atomicAdd(out + 1, v.y);
  atomicAdd(out + 2, v.z);
  atomicAdd(out + 3, v.w);
  if (g == 0) atomicAdd(DEG + d, 1.0f);
}

// ---------------------------------------------------------------------------
// Dense SAGE update via V_WMMA_F32_16X16X4_F32:
//   Y = tanh( (MSG/deg) @ Wl + X @ Wr + b )
// A layout (f32 16x4): lane = M (mod 16); VGPR0 = K={0|2}, VGPR1 = K={1|3}
//   (high half-wave holds K+2). B mirrors with lane = N.
// C/D layout: lane = N (mod 16), VGPR v = M = v + 8*(lane>=16).
// Row indices are clamped (not predicated) so EXEC stays all-ones for WMMA.
// ---------------------------------------------------------------------------
__global__ void __launch_bounds__(256)
sage_dense_wmma(const float* __restrict__ X, const float* __restrict__ MSG,
                const float* __restrict__ DEG,
                const float* __restrict__ Wl, const float* __restrict__ Wr,
                const float* __restrict__ bias,
                float* __restrict__ Y, int n) {
  __shared__ float sWl[LW * LW];
  __shared__ float sWr[LW * LW];
  __shared__ float sB[LW];
  for (int i = threadIdx.x; i < LW * LW; i += blockDim.x) {
    sWl[i] = Wl[i];
    sWr[i] = Wr[i];
  }
  if (threadIdx.x < LW) sB[threadIdx.x] = bias[threadIdx.x];
  __syncthreads();

  const int wave  = threadIdx.x >> 5;
  const int lane  = threadIdx.x & 31;
  const int m16   = lane & 15;       // M (A operand) / N (B,C,D operands)
  const int hi    = lane >> 4;       // half-wave
  const int khalf = hi << 1;         // K offset 0 or 2
  const int mOff  = hi << 3;         // +8 rows in C/D for high half

  const int tiles = (n + 15) >> 4;
  for (int t = blockIdx.x * WPB + wave; t < tiles; t += gridDim.x * WPB) {
    const int rowbase = t << 4;
    int row = rowbase + m16;
    if (row >= n) row = n - 1;       // clamp: keep EXEC full

    const float invd = 1.0f / fmaxf(DEG[row], 1.0f);

    v2f amean[8], ax[8];
#pragma unroll
    for (int s = 0; s < 8; ++s) {
      const int k = s * 4 + khalf;
      v2f mm = *(const v2f*)(MSG + (size_t)row * LW + k);
      amean[s] = mm * invd;
      ax[s] = *(const v2f*)(X + (size_t)row * LW + k);
    }

#pragma unroll
    for (int j = 0; j < 2; ++j) {          // two 16-wide output column tiles
      const int col = j * 16 + m16;
      const float bv = sB[col];
      v8f acc = {bv, bv, bv, bv, bv, bv, bv, bv};   // bias broadcast over rows
#pragma unroll
      for (int s = 0; s < 8; ++s) {        // mean @ Wl
        const int kb = s * 4 + khalf;
        v2f bf;
        bf.x = sWl[kb * LW + col];
        bf.y = sWl[(kb + 1) * LW + col];
        acc = __builtin_amdgcn_wmma_f32_16x16x4_f32(
            false, amean[s], false, bf, (short)0, acc, false, false);
      }
#pragma unroll
      for (int s = 0; s < 8; ++s) {        // x @ Wr
        const int kb = s * 4 + khalf;
        v2f bf;
        bf.x = sWr[kb * LW + col];
        bf.y = sWr[(kb + 1) * LW + col];
        acc = __builtin_amdgcn_wmma_f32_16x16x4_f32(
            false, ax[s], false, bf, (short)0, acc, false, false);
      }
#pragma unroll
      for (int v = 0; v < 8; ++v) {
        const int r = rowbase + v + mOff;
        if (r < n) Y[(size_t)r * LW + col] = tanhf(acc[v]);
      }
    }
  }
}

// ---------------------------------------------------------------------------
// Fused trailing MLP: 32 -> 16 -> 16 -> 16 -> 2 (tanh on hidden layers)
// ---------------------------------------------------------------------------
__global__ void __launch_bounds__(256)
mlp_kernel(const float* __restrict__ X,
           const float* __restrict__ W1, const float* __restrict__ b1,
           const float* __restrict__ W2, const float* __restrict__ b2,
           const float* __restrict__ W3, const float* __restrict__ b3,
           const float* __restrict__ Wf, const float* __restrict__ bf,
           float* __restrict__ OUT2, int n) {
  __shared__ float sW1[LW * 16], sW2[16 * 16], sW3[16 * 16], sWf[16 * 2];
  __shared__ float sb1[16], sb2[16], sb3[16], sbf[2];
  for (int i = threadIdx.x; i < LW * 16; i += blockDim.x) sW1[i] = W1[i];
  for (int i = threadIdx.x; i < 256; i += blockDim.x) { sW2[i] = W2[i]; sW3[i] = W3[i]; }
  for (int i = threadIdx.x; i < 32; i += blockDim.x) sWf[i] = Wf[i];
  if (threadIdx.x < 16) {
    sb1[threadIdx.x] = b1[threadIdx.x];
    sb2[threadIdx.x] = b2[threadIdx.x];
    sb3[threadIdx.x] = b3[threadIdx.x];
  }
  if (threadIdx.x < 2) sbf[threadIdx.x] = bf[threadIdx.x];
  __syncthreads();

  int i = blockIdx.x * blockDim.x + threadIdx.x;
  if (i >= n) return;

  float h0[LW];
#pragma unroll
  for (int c = 0; c < LW; c += 4) {
    float4 v = *(const float4*)(X + (size_t)i * LW + c);
    h0[c] = v.x; h0[c + 1] = v.y; h0[c + 2] = v.z; h0[c + 3] = v.w;
  }
  float h1[16], h2[16], h3[16];
#pragma unroll
  for (int j = 0; j < 16; ++j) {
    float a = sb1[j];
#pragma unroll
    for (int c = 0; c < LW; ++c) a += h0[c] * sW1[c * 16 + j];
    h1[j] = tanhf(a);
  }
#pragma unroll
  for (int j = 0; j < 16; ++j) {
    float a = sb2[j];
#pragma unroll
    for (int c = 0; c < 16; ++c) a += h1[c] * sW2[c * 16 + j];
    h2[j] = tanhf(a);
  }
#pragma unroll
  for (int j = 0; j < 16; ++j) {
    float a = sb3[j];
#pragma unroll
    for (int c = 0; c < 16; ++c) a += h2[c] * sW3[c * 16 + j];
    h3[j] = tanhf(a);
  }
  float y0 = sbf[0], y1 = sbf[1];
#pragma unroll
  for (int c = 0; c < 16; ++c) {
    y0 += h3[c] * sWf[c * 2 + 0];
    y1 += h3[c] * sWf[c * 2 + 1];
  }
  OUT2[(size_t)i * 2 + 0] = y0;
  OUT2[(size_t)i * 2 + 1] = y1;
}

// ---------------------------------------------------------------------------
// QR of an [n,2] matrix via its 2x2 Gram matrix (Gram-Schmidt Q)
// ---------------------------------------------------------------------------
__global__ void gram_reduce(const float* __restrict__ OUT2,
                            float* __restrict__ RED, int n) {
  __shared__ float s0[256], s1[256], s2[256];
  float a00 = 0.f, a01 = 0.f, a11 = 0.f;
  for (int i = blockIdx.x * blockDim.x + threadIdx.x; i < n;
       i += gridDim.x * blockDim.x) {
    float a = OUT2[(size_t)i * 2 + 0];
    float b = OUT2[(size_t)i * 2 + 1];
    a00 += a * a; a01 += a * b; a11 += b * b;
  }
  int t = threadIdx.x;
  s0[t] = a00; s1[t] = a01; s2[t] = a11;
  __syncthreads();
  for (int off = 128; off > 0; off >>= 1) {
    if (t < off) { s0[t] += s0[t + off]; s1[t] += s1[t + off]; s2[t] += s2[t + off]; }
    __syncthreads();
  }
  if (t == 0) {
    atomicAdd(RED + 0, s0[0]);
    atomicAdd(RED + 1, s1[0]);
    atomicAdd(RED + 2, s2[0]);
  }
}

__global__ void qr_finalize(float* __restrict__ RED) {
  if (threadIdx.x == 0 && blockIdx.x == 0) {
    float s00 = RED[0], s01 = RED[1], s11 = RED[2];
    float s00c = fmaxf(s00, 1e-30f);
    float inv_r11 = rsqrtf(s00c);
    float k = s01 / s00c;                    // r12 / r11
    float r22sq = s11 - s01 * k;             // ||c1 - proj||^2
    float inv_r22 = rsqrtf(fmaxf(r22sq, 1e-30f));
    RED[4] = inv_r11; RED[5] = k; RED[6] = inv_r22;
  }
}

__global__ void q_write(const float* __restrict__ OUT2,
                        const float* __restrict__ RED,
                        float* __restrict__ Q, int n) {
  int i = blockIdx.x * blockDim.x + threadIdx.x;
  if (i >= n) return;
  float inv_r11 = RED[4], k = RED[5], inv_r22 = RED[6];
  float a = OUT2[(size_t)i * 2 + 0];
  float b = OUT2[(size_t)i * 2 + 1];
  Q[(size_t)i * 2 + 0] = a * inv_r11;
  Q[(size_t)i * 2 + 1] = (b - k * a) * inv_r22;
}

// ---------------------------------------------------------------------------
static inline int cdiv(long a, long b) { return (int)((a + b - 1) / b); }

extern "C" void kernel_launch(void* const* d_in, const int* in_sizes, int n_in,
                              void* d_out, int out_size, void* d_ws, size_t ws_size,
                              hipStream_t stream) {
  const int NSv[5] = {200000, 40000, 4000, 100, 2};
  const int* src[4] = {(const int*)d_in[0], (const int*)d_in[3],
                       (const int*)d_in[6], (const int*)d_in[9]};
  const int* dst[4] = {(const int*)d_in[1], (const int*)d_in[4],
                       (const int*)d_in[7], (const int*)d_in[10]};
  const int* inv[4] = {(const int*)d_in[2], (const int*)d_in[5],
                       (const int*)d_in[8], (const int*)d_in[11]};
  const float* Wc_l = (const float*)d_in[12];
  const float* Wc_r = (const float*)d_in[13];
  const float* bc   = (const float*)d_in[14];
  const float* Wp_l = (const float*)d_in[15];
  const float* Wp_r = (const float*)d_in[16];
  const float* bp   = (const float*)d_in[17];
  const float* W1 = (const float*)d_in[18]; const float* b1 = (const float*)d_in[19];
  const float* W2 = (const float*)d_in[20]; const float* b2 = (const float*)d_in[21];
  const float* W3 = (const float*)d_in[22]; const float* b3 = (const float*)d_in[23];
  const float* Wf = (const float*)d_in[24]; const float* bf = (const float*)d_in[25];

  float* ws = (float*)d_ws;
  const size_t NMAX = 200000;
  float* X   = ws;                  // [NMAX, 32] ping
  float* Y   = ws + NMAX * 32;      // [NMAX, 32] pong
  float* MSG = ws + 2 * NMAX * 32;  // [NMAX, 32] aggregation scratch
  float* DEG = ws + 3 * NMAX * 32;  // [NMAX]
  float* RED = DEG + NMAX;          // [8] gram sums + QR factors

  coarse_kernel<<<1, 64, 0, stream>>>(Wc_l, Wc_r, bc, X);

  for (int k = 3; k >= 0; --k) {
    const int n = NSv[k];
    const int E = n * 16;
    unpool_kernel<<<cdiv((long)n * 8, 256), 256, 0, stream>>>(Y, X, inv[k], n);
    { float* t = X; X = Y; Y = t; }
    for (int p = 0; p < 2; ++p) {
      zero_kernel<<<cdiv((long)n * 32, 256), 256, 0, stream>>>(MSG, n * 32);
      zero_kernel<<<cdiv(n, 256), 256, 0, stream>>>(DEG, n);
      aggregate_kernel<<<cdiv((long)E * 8, 256), 256, 0, stream>>>(
          X, src[k], dst[k], MSG, DEG, E);
      const int tiles = (n + 15) >> 4;
      sage_dense_wmma<<<cdiv(tiles, WPB), 256, 0, stream>>>(
          X, MSG, DEG, Wp_l + p * LW * LW, Wp_r + p * LW * LW, bp + p * LW, Y, n);
      { float* t = X; X = Y; Y = t; }
    }
  }

  const int n0 = NSv[0];
  float* OUT2 = Y;  // reuse pong buffer for [n0, 2]
  mlp_kernel<<<cdiv(n0, 256), 256, 0, stream>>>(X, W1, b1, W2, b2, W3, b3,
                                                Wf, bf, OUT2, n0);
  zero_kernel<<<1, 32, 0, stream>>>(RED, 8);
  gram_reduce<<<256, 256, 0, stream>>>(OUT2, RED, n0);
  qr_finalize<<<1, 1, 0, stream>>>(RED);
  q_write<<<cdiv(n0, 256), 256, 0, stream>>>(OUT2, RED, (float*)d_out, n0);
}